// OutputBlock_43465069035926
// MI455X (gfx1250) — compile-verified
//
#include <hip/hip_runtime.h>
#include <hip/hip_bf16.h>

typedef __attribute__((ext_vector_type(2))) float v2f;
typedef __attribute__((ext_vector_type(4))) float v4f;
typedef __attribute__((ext_vector_type(8))) float v8f;

// Problem constants (fixed by the reference)
#define A_TOT   2048   // N_ATOMS
#define EMB_K   128    // EMB
#define NCOL    46     // 2*S_EXP + 5*P_EXP
#define NPAD    48     // padded coeff stride
#define NMOL    64
#define APM     32     // atoms per molecule (uniform N)

// ---------------------------------------------------------------------------
// Kernel 1: coeffs[2048,46] = out[2048,128] @ W[128,46] + b   via f32 WMMA.
// Block = 128 threads (4 waves). Each wave owns one 16x16 tile of coeffs.
// Grid = (2048/16/4, ceil(46/16)) = (32, 3).
// ---------------------------------------------------------------------------
__global__ __launch_bounds__(128) void coeff_gemm_wmma(
    const float* __restrict__ A,     // [2048,128] row-major
    const float* __restrict__ W,     // [128,46]   row-major
    const float* __restrict__ bias,  // [46]
    float* __restrict__ C)           // [2048,48]  padded scratch
{
    const int wave = threadIdx.x >> 5;
    const int lane = threadIdx.x & 31;

    const int mtile = blockIdx.x * 4 + wave;
    const int m0    = mtile * 16;
    const int n0    = blockIdx.y * 16;

    // A-matrix 16x4 f32 layout: lanes 0-15 -> rows M=0..15 with K=kb..kb+1,
    // lanes 16-31 -> same rows with K=kb+2..kb+3.
    const int rowm = lane & 15;          // M within tile / N within tile
    const int kb   = (lane >> 4) << 1;   // 0 or 2

    const int  col  = n0 + rowm;                       // global N for B & C
    const float msk = (col < NCOL) ? 1.0f : 0.0f;      // zero-pad cols 46,47
    const int  colc = (col < NCOL) ? col : (NCOL - 1); // clamped (no OOB read)

    const float* __restrict__ arow = A + (m0 + rowm) * EMB_K;

    v8f acc = {};   // 16x16 f32 accumulator, 8 VGPRs
    #pragma unroll
    for (int k0 = 0; k0 < EMB_K; k0 += 4) {
        // A fragment: two consecutive K values -> one 8B load
        v2f a = *(const v2f*)(arow + k0 + kb);
        // B fragment: rows k0+kb, k0+kb+1 of W at column `col`
        v2f b;
        b.x = W[(k0 + kb)     * NCOL + colc] * msk;
        b.y = W[(k0 + kb + 1) * NCOL + colc] * msk;
        // D = A*B + C  (args: neg_a, A, neg_b, B, c_mod, C, reuse_a, reuse_b)
        acc = __builtin_amdgcn_wmma_f32_16x16x4_f32(
                  false, a, false, b, (short)0, acc, false, false);
    }

    const float bv = bias[colc] * msk;

    // C/D layout: VGPR r -> M = r + 8*(lane>=16), N = lane&15
    const int mbase = m0 + ((lane >> 4) << 3);
    #pragma unroll
    for (int r = 0; r < 8; ++r) {
        C[(mbase + r) * NPAD + col] = acc[r] + bv;   // col < 48 always
    }
}

// ---------------------------------------------------------------------------
// Kernel 2: fused gaussians + angular term + segment reduction + broadcast.
// One 256-thread block per molecule m. Columns collapse 2048 -> 64 because
// dist/ang depend on j only through coords[j/32].
// ---------------------------------------------------------------------------
__global__ __launch_bounds__(256) void mol_field_kernel(
    const float* __restrict__ coeffs,  // [2048,48] padded
    const float* __restrict__ R,       // [2048,3]
    const float* __restrict__ coords,  // [64,3]
    float* __restrict__ outp)          // [64,2048]
{
    __shared__ float scoef[APM][NPAD];   // 6 KB: coeff rows of this mol's atoms
    __shared__ float sR[APM][3];
    __shared__ float scrd[NMOL][3];
    __shared__ float partial[4][NMOL];
    __shared__ float sres[NMOL];

    const int m = blockIdx.x;
    const int t = threadIdx.x;

    // Stage atom coefficients (rows m*32 .. m*32+31 are contiguous, stride 48)
    for (int idx = t; idx < APM * NPAD; idx += 256)
        (&scoef[0][0])[idx] = coeffs[m * APM * NPAD + idx];
    for (int idx = t; idx < APM * 3; idx += 256)
        (&sR[0][0])[idx] = R[m * APM * 3 + idx];
    for (int idx = t; idx < NMOL * 3; idx += 256)
        (&scrd[0][0])[idx] = coords[idx];
    __syncthreads();

    const int c   = t & 63;   // molecule-coord column
    const int grp = t >> 6;   // atom group 0..3 (8 atoms each)

    const float cx = scrd[c][0], cy = scrd[c][1], cz = scrd[c][2];

    float acc = 0.0f;
    #pragma unroll
    for (int ii = 0; ii < 8; ++ii) {
        const int i = grp * 8 + ii;
        const float dx = sR[i][0] - cx;
        const float dy = sR[i][1] - cy;
        const float dz = sR[i][2] - cz;
        const float adx = fabsf(dx), ady = fabsf(dy), adz = fabsf(dz);
        const float d2 = dx * dx + dy * dy + dz * dz;
        const float d  = sqrtf(d2);

        const float* cf = scoef[i];   // broadcast reads (same addr per group)

        // s-type: sum_e s_c[e] * exp(-(s_a[e]^2) * dist)
        #pragma unroll
        for (int e = 0; e < 8; ++e) {
            float al = cf[e];
            al *= al;
            acc += cf[8 + e] * __expf(-al * d);
        }
        // p-type: sum_e p_c[e] * exp(-(p_a[e]^2) * dist^2) * (v . |diff|)
        #pragma unroll
        for (int e = 0; e < 6; ++e) {
            float pa = cf[16 + e];
            pa *= pa;
            const float pc  = cf[22 + e];
            const float ang = cf[28 + 3 * e] * adx
                            + cf[29 + 3 * e] * ady
                            + cf[30 + 3 * e] * adz;
            acc += pc * __expf(-pa * d2) * ang;
        }
    }

    partial[grp][c] = acc;
    __syncthreads();

    if (t < NMOL)
        sres[t] = partial[0][t] + partial[1][t] + partial[2][t] + partial[3][t];
    __syncthreads();

    // out[m, j] = sres[j >> 5]; thread t writes j = 8t .. 8t+7, and
    // (8t..8t+7)>>5 == t>>2 is constant per thread -> two 128-bit stores.
    const float v = sres[t >> 2];
    const v4f vv = {v, v, v, v};
    v4f* dst = (v4f*)(outp + m * 2048 + t * 8);
    dst[0] = vv;
    dst[1] = vv;
}

// ---------------------------------------------------------------------------
extern "C" void kernel_launch(void* const* d_in, const int* in_sizes, int n_in,
                              void* d_out, int out_size, void* d_ws, size_t ws_size,
                              hipStream_t stream) {
    const float* out_emb = (const float*)d_in[0];  // (2048,128) f32
    // d_in[1] = Z (unused by the math), d_in[4] = N (uniform 32)
    const float* R       = (const float*)d_in[2];  // (2048,3)
    const float* coords  = (const float*)d_in[3];  // (64,3)
    const float* W       = (const float*)d_in[5];  // (128,46)
    const float* bias    = (const float*)d_in[6];  // (46,)

    float* coeffs = (float*)d_ws;   // 2048*48*4 = 384 KB scratch
    float* outp   = (float*)d_out;  // (64,2048) f32

    dim3 g1(A_TOT / 16 / 4, (NCOL + 15) / 16);  // (32, 3)
    coeff_gemm_wmma<<<g1, dim3(128), 0, stream>>>(out_emb, W, bias, coeffs);

    mol_field_kernel<<<dim3(NMOL), dim3(256), 0, stream>>>(coeffs, R, coords, outp);
}